// MPDWConv_16561393894113
// MI455X (gfx1250) — compile-verified
//
#include <hip/hip_runtime.h>

typedef __attribute__((ext_vector_type(2))) float v2f;
typedef __attribute__((ext_vector_type(8))) float v8f;

#define Bn 8
#define En 256
#define Hn 128
#define Wn 128
#define HW (Hn * Wn)        // 16384
#define CHW (En * HW)       // 4194304

// ---------------------------------------------------------------------------
// Kernel 1: fused depthwise chain.
//   x0 = dw3x3(x, w0, b0, d=1, p=1)             (kept in LDS only)
//   xc[:, 0:64]    = x0[:, 0:64]
//   xc[:, 64:128]  = dw3x3(x0[:, 64:128],  w1, b1, d=3, p=3)
//   xc[:, 128:192] = dw5x5(x0[:,128:192],  w2, b2, d=3, p=6)
//   xc[:, 192:256] = dw5x5(x0[:,192:256],  w2, b2, d=3, p=6)   (w2 reused)
// One block per (b*c, 32x32 tile). Halo: branch conv needs x0 +-6, x0 needs
// x +-1 more => x halo 7.
// ---------------------------------------------------------------------------
__global__ __launch_bounds__(256) void dw_fused_kernel(
    const float* __restrict__ x,
    const float* __restrict__ w0, const float* __restrict__ b0,
    const float* __restrict__ w1, const float* __restrict__ b1,
    const float* __restrict__ w2, const float* __restrict__ b2,
    float* __restrict__ xc)
{
    __shared__ float xs[46 * 46];    // x tile, halo 7
    __shared__ float x0s[44 * 44];   // x0 tile, halo 6

    const int bc  = blockIdx.x;          // b*256 + c
    const int c   = bc & 255;
    const int ty  = blockIdx.y * 32;
    const int tx  = blockIdx.z * 32;
    const int tid = threadIdx.x;

    const float* xp = x + (size_t)bc * HW;

    // ---- stage x tile into LDS (zero-padded outside image) ----
    for (int idx = tid; idx < 46 * 46; idx += 256) {
        int r  = idx / 46, q = idx % 46;
        int gh = ty - 7 + r, gw = tx - 7 + q;
        float v = 0.f;
        if (gh >= 0 && gh < Hn && gw >= 0 && gw < Wn)
            v = xp[gh * Wn + gw];
        xs[idx] = v;
    }

    // stage-1 weights are uniform per block (c is SGPR-derived)
    float w0c[9];
#pragma unroll
    for (int i = 0; i < 9; ++i) w0c[i] = w0[c * 9 + i];
    const float b0c = b0[c];
    __syncthreads();

    // ---- compute x0 tile (halo 6). Out-of-image positions are ZERO,
    //      because the branch convs zero-pad x0 (not b0-biased values). ----
    for (int idx = tid; idx < 44 * 44; idx += 256) {
        int r  = idx / 44, q = idx % 44;
        int gh = ty - 6 + r, gw = tx - 6 + q;
        float v = 0.f;
        if (gh >= 0 && gh < Hn && gw >= 0 && gw < Wn) {
            float acc = b0c;
#pragma unroll
            for (int ky = 0; ky < 3; ++ky)
#pragma unroll
                for (int kx = 0; kx < 3; ++kx)
                    acc += w0c[ky * 3 + kx] * xs[(r + ky) * 46 + (q + kx)];
            v = acc;
        }
        x0s[idx] = v;
    }
    __syncthreads();

    float* op = xc + (size_t)bc * HW;

    if (c < 64) {
        // passthrough quarter
        for (int idx = tid; idx < 1024; idx += 256) {
            int oh = idx >> 5, ow = idx & 31;
            op[(ty + oh) * Wn + (tx + ow)] = x0s[(oh + 6) * 44 + (ow + 6)];
        }
    } else if (c < 128) {
        // 3x3, dilation 3, padding 3: taps at x0(o + 3k - 3), k=0..2
        const int cw = c - 64;
        float wk[9];
#pragma unroll
        for (int i = 0; i < 9; ++i) wk[i] = w1[cw * 9 + i];
        const float bb = b1[cw];
        for (int idx = tid; idx < 1024; idx += 256) {
            int oh = idx >> 5, ow = idx & 31;
            float acc = bb;
#pragma unroll
            for (int ky = 0; ky < 3; ++ky)
#pragma unroll
                for (int kx = 0; kx < 3; ++kx)
                    acc += wk[ky * 3 + kx] *
                           x0s[(oh + 3 + 3 * ky) * 44 + (ow + 3 + 3 * kx)];
            op[(ty + oh) * Wn + (tx + ow)] = acc;
        }
    } else {
        // 5x5, dilation 3, padding 6: taps at x0(o + 3k - 6), k=0..4
        // w2/b2 reused for BOTH upper quarters: weight channel = (c-128)&63
        const int cw = (c - 128) & 63;
        float wk[25];
#pragma unroll
        for (int i = 0; i < 25; ++i) wk[i] = w2[cw * 25 + i];
        const float bb = b2[cw];
        for (int idx = tid; idx < 1024; idx += 256) {
            int oh = idx >> 5, ow = idx & 31;
            float acc = bb;
#pragma unroll
            for (int ky = 0; ky < 5; ++ky)
#pragma unroll
                for (int kx = 0; kx < 5; ++kx)
                    acc += wk[ky * 5 + kx] *
                           x0s[(oh + 3 * ky) * 44 + (ow + 3 * kx)];
            op[(ty + oh) * Wn + (tx + ow)] = acc;
        }
    }
}

// ---------------------------------------------------------------------------
// Kernel 2: pointwise conv as channel GEMM with V_WMMA_F32_16X16X4_F32.
//   y[b,o,p] = sum_c w_pw[o,c] * xc[b,c,p] + b_pw[o]
// Block = 512 threads = 16 waves. Block owns one 16-pixel N-tile; XC[256x16]
// staged in LDS once and shared by all 16 waves; wave w owns M-tile o=[16w,
// 16w+16). K=256 swept with 64 WMMA 16x16x4 ops.
//
// f32 WMMA layouts (wave32):
//   A 16x4:  VGPR j, lane L: M = L%16, K = j + 2*(L/16)  -> 8B load from w_pw
//   B 4x16:  VGPR j, lane L: N = L%16, K = j + 2*(L/16)  -> LDS reads
//   C/D:     VGPR r, lane L: M = r + 8*(L/16), N = L%16
// ---------------------------------------------------------------------------
__global__ __launch_bounds__(512) void pw_wmma_kernel(
    const float* __restrict__ xc,
    const float* __restrict__ wpw,
    const float* __restrict__ bpw,
    float* __restrict__ y)
{
    __shared__ float bs[256 * 16];   // XC[c][p] tile, 16 KB

    const int tid = threadIdx.x;
    const int blk = blockIdx.x;
    const int b   = blk >> 10;              // 1024 pixel-tiles per batch
    const int p0  = (blk & 1023) << 4;

    const float* xp = xc + (size_t)b * CHW + p0;
    {
        const int p  = tid & 15;
        const int c0 = tid >> 4;            // 0..31
#pragma unroll
        for (int i = 0; i < 8; ++i) {
            const int cc = c0 + i * 32;
            bs[cc * 16 + p] = xp[(size_t)cc * HW + p];
        }
    }
    __syncthreads();

    const int wave = tid >> 5;              // 0..15
    const int lane = tid & 31;
    const int lm   = lane & 15;
    const int lh   = lane >> 4;
    const int o0   = wave << 4;

    v8f acc = {};
    // A pointer: row (o0+lm) of w_pw, pre-offset by lane-half K shift.
    const float* arow = wpw + (size_t)(o0 + lm) * 256 + 2 * lh;
    const float* brow = bs + 2 * lh * 16 + lm;

#pragma unroll 8
    for (int k0 = 0; k0 < 256; k0 += 4) {
        v2f a = *(const v2f*)(arow + k0);             // K = k0+2h, k0+2h+1
        v2f bfrag;
        bfrag.x = brow[k0 * 16];                      // K = k0   + 2h
        bfrag.y = brow[(k0 + 1) * 16];                // K = k0+1 + 2h
        acc = __builtin_amdgcn_wmma_f32_16x16x4_f32(
            /*neg_a=*/false, a, /*neg_b=*/false, bfrag,
            /*c_mod=*/(short)0, acc, /*reuse_a=*/false, /*reuse_b=*/false);
    }

    float* yp = y + (size_t)b * CHW + p0 + lm;
#pragma unroll
    for (int r = 0; r < 8; ++r) {
        const int o = o0 + r + 8 * lh;
        yp[(size_t)o * HW] = acc[r] + bpw[o];
    }
}

// ---------------------------------------------------------------------------
extern "C" void kernel_launch(void* const* d_in, const int* in_sizes, int n_in,
                              void* d_out, int out_size, void* d_ws, size_t ws_size,
                              hipStream_t stream) {
    const float* x   = (const float*)d_in[0];
    const float* w0  = (const float*)d_in[1];
    const float* b0  = (const float*)d_in[2];
    const float* w1  = (const float*)d_in[3];
    const float* b1  = (const float*)d_in[4];
    const float* w2  = (const float*)d_in[5];
    const float* b2  = (const float*)d_in[6];
    const float* wpw = (const float*)d_in[7];
    const float* bpw = (const float*)d_in[8];

    float* xc = (float*)d_ws;   // B*E*H*W fp32 intermediate (concat result)

    dim3 g1(Bn * En, Hn / 32, Wn / 32);
    dw_fused_kernel<<<g1, 256, 0, stream>>>(x, w0, b0, w1, b1, w2, b2, xc);

    pw_wmma_kernel<<<Bn * (HW / 16), 512, 0, stream>>>(xc, wpw, bpw, (float*)d_out);
}